// MultiInputLSTMCell_5574867550454
// MI455X (gfx1250) — compile-verified
//
#include <hip/hip_runtime.h>
#include <hip/hip_bf16.h>

#define H      2048
#define H3     6144
#define KDIM   2048
#define CNUM   64
#define NSLICE 64
#define KSLICE 32      // KDIM / NSLICE
#define KC     128     // K-chunk for WMMA GEMM
#define LDA    132     // padded LDS row stride (banks: 132 % 64 = 4 -> conflict-free)

typedef __attribute__((ext_vector_type(2))) float v2f;
typedef __attribute__((ext_vector_type(8))) float v8f;

// ---------------------------------------------------------------------------
// Kernel 1: partial sums for pre = h0 @ W_hh + x @ W_ih   (K split in 64 slices)
// grid = (6144/1024, 64), block = 256.  Each thread: 4 consecutive columns.
// ---------------------------------------------------------------------------
__global__ void __launch_bounds__(256) gate_matvec_partial(
    const float* __restrict__ x, const float* __restrict__ h0,
    const float* __restrict__ Wih, const float* __restrict__ Whh,
    float* __restrict__ preP) {
  __shared__ float xs[KSLICE];
  __shared__ float hs[KSLICE];
  const int tid = threadIdx.x;
  const int s   = blockIdx.y;
  const int k0  = s * KSLICE;
  if (tid < KSLICE) { xs[tid] = x[k0 + tid]; hs[tid] = h0[k0 + tid]; }
  __syncthreads();

  const int col = (blockIdx.x * 256 + tid) * 4;
  float4 acc = make_float4(0.f, 0.f, 0.f, 0.f);
  #pragma unroll 8
  for (int kk = 0; kk < KSLICE; ++kk) {
    const float xv = xs[kk];
    const float hv = hs[kk];
    const float4 wi = *(const float4*)(Wih + (size_t)(k0 + kk) * H3 + col);
    const float4 wh = *(const float4*)(Whh + (size_t)(k0 + kk) * H3 + col);
    acc.x += xv * wi.x + hv * wh.x;
    acc.y += xv * wi.y + hv * wh.y;
    acc.z += xv * wi.z + hv * wh.z;
    acc.w += xv * wi.w + hv * wh.w;
  }
  *(float4*)(preP + (size_t)s * H3 + col) = acc;
}

// ---------------------------------------------------------------------------
// Kernel 2: partial sums for alpha_wi = x @ aW_ih.  grid = (2048/1024, 64).
// ---------------------------------------------------------------------------
__global__ void __launch_bounds__(256) alpha_matvec_partial(
    const float* __restrict__ x, const float* __restrict__ Wa,
    float* __restrict__ awiP) {
  __shared__ float xs[KSLICE];
  const int tid = threadIdx.x;
  const int s   = blockIdx.y;
  const int k0  = s * KSLICE;
  if (tid < KSLICE) xs[tid] = x[k0 + tid];
  __syncthreads();

  const int col = (blockIdx.x * 256 + tid) * 4;
  float4 acc = make_float4(0.f, 0.f, 0.f, 0.f);
  #pragma unroll 8
  for (int kk = 0; kk < KSLICE; ++kk) {
    const float xv = xs[kk];
    const float4 w = *(const float4*)(Wa + (size_t)(k0 + kk) * H + col);
    acc.x += xv * w.x;
    acc.y += xv * w.y;
    acc.z += xv * w.z;
    acc.w += xv * w.w;
  }
  *(float4*)(awiP + (size_t)s * H + col) = acc;
}

// ---------------------------------------------------------------------------
// Kernel 3: alpha_wh = c_input(64x2048) @ aW_hh(2048x2048) via v_wmma_f32_16x16x4_f32.
// grid = (32, 4), block = 128 (4 waves).  Wave -> one 16x16 tile, full K loop.
// A chunk (16 x 128) staged in LDS with padded stride; B streamed from global/L2.
// A layout (ISA 7.12.2): lane 0-15 row M, VGPR0 = K{0|2}, VGPR1 = K{1|3} (low|high half).
// B layout (row striped across lanes): VGPR0 = K{0|2} row, VGPR1 = K{1|3} row.
// 4 independent accumulator chains to hide WMMA latency.
// ---------------------------------------------------------------------------
__global__ void __launch_bounds__(128) alpha_gemm_wmma(
    const float* __restrict__ A,      // c_input, 64 x 2048 row-major
    const float* __restrict__ B,      // alpha_weight_hh, 2048 x 2048 row-major
    float* __restrict__ Dst) {        // alpha_wh, 64 x 2048
  __shared__ float Alds[16 * LDA];
  const int tid  = threadIdx.x;
  const int wave = tid >> 5;
  const int lane = tid & 31;
  const int half = lane >> 4;
  const int l    = lane & 15;
  const int m0   = blockIdx.y * 16;
  const int n0   = (blockIdx.x * 4 + wave) * 16;

  v8f acc0 = {};
  v8f acc1 = {};
  v8f acc2 = {};
  v8f acc3 = {};

  for (int kc0 = 0; kc0 < KDIM; kc0 += KC) {
    __syncthreads();
    // cooperative load of A[m0:m0+16, kc0:kc0+128] -> LDS (coalesced)
    for (int i = tid; i < 16 * KC; i += 128) {
      const int r  = i >> 7;     // / KC
      const int kk = i & (KC - 1);
      Alds[r * LDA + kk] = A[(size_t)(m0 + r) * KDIM + kc0 + kk];
    }
    __syncthreads();

    const float* arow = &Alds[l * LDA];
    #pragma unroll
    for (int kk = 0; kk < KC; kk += 16) {
      {
        const int k = kk;
        v2f a; a.x = arow[k + 2 * half]; a.y = arow[k + 2 * half + 1];
        const float* bp = B + (size_t)(kc0 + k + 2 * half) * H + n0 + l;
        v2f b; b.x = bp[0]; b.y = bp[H];
        acc0 = __builtin_amdgcn_wmma_f32_16x16x4_f32(
            false, a, false, b, (short)0, acc0, false, false);
      }
      {
        const int k = kk + 4;
        v2f a; a.x = arow[k + 2 * half]; a.y = arow[k + 2 * half + 1];
        const float* bp = B + (size_t)(kc0 + k + 2 * half) * H + n0 + l;
        v2f b; b.x = bp[0]; b.y = bp[H];
        acc1 = __builtin_amdgcn_wmma_f32_16x16x4_f32(
            false, a, false, b, (short)0, acc1, false, false);
      }
      {
        const int k = kk + 8;
        v2f a; a.x = arow[k + 2 * half]; a.y = arow[k + 2 * half + 1];
        const float* bp = B + (size_t)(kc0 + k + 2 * half) * H + n0 + l;
        v2f b; b.x = bp[0]; b.y = bp[H];
        acc2 = __builtin_amdgcn_wmma_f32_16x16x4_f32(
            false, a, false, b, (short)0, acc2, false, false);
      }
      {
        const int k = kk + 12;
        v2f a; a.x = arow[k + 2 * half]; a.y = arow[k + 2 * half + 1];
        const float* bp = B + (size_t)(kc0 + k + 2 * half) * H + n0 + l;
        v2f b; b.x = bp[0]; b.y = bp[H];
        acc3 = __builtin_amdgcn_wmma_f32_16x16x4_f32(
            false, a, false, b, (short)0, acc3, false, false);
      }
    }
  }

  // D layout: VGPR r -> row (r + 8*half), col = lane&15
  #pragma unroll
  for (int r = 0; r < 8; ++r) {
    const int row = m0 + r + 8 * half;
    Dst[(size_t)row * H + n0 + l] = (acc0[r] + acc1[r]) + (acc2[r] + acc3[r]);
  }
}

// ---------------------------------------------------------------------------
// Kernel 4: reduce K-slices, add biases.  8192 threads total.
// ---------------------------------------------------------------------------
__global__ void __launch_bounds__(256) finalize_vec(
    const float* __restrict__ preP, const float* __restrict__ awiP,
    const float* __restrict__ bias, const float* __restrict__ abias,
    float* __restrict__ pre, float* __restrict__ awi) {
  const int j = blockIdx.x * 256 + threadIdx.x;
  if (j < H3) {
    float a = bias[j];
    #pragma unroll 8
    for (int s = 0; s < NSLICE; ++s) a += preP[(size_t)s * H3 + j];
    pre[j] = a;
  } else {
    const int j2 = j - H3;
    float a = abias[j2];
    #pragma unroll 8
    for (int s = 0; s < NSLICE; ++s) a += awiP[(size_t)s * H + j2];
    awi[j2] = a;
  }
}

// ---------------------------------------------------------------------------
// Kernel 5: gates + exp-normalized merge + outputs.  One thread per hidden unit.
// ---------------------------------------------------------------------------
__global__ void __launch_bounds__(256) fuse_out(
    const float* __restrict__ pre, const float* __restrict__ awi,
    const float* __restrict__ awh, const float* __restrict__ cinp,
    float* __restrict__ out) {
  const int h = blockIdx.x * 256 + threadIdx.x;
  const float iv = pre[h];
  const float ov = pre[H + h];
  const float gv = pre[2 * H + h];
  const float ig = 1.f / (1.f + expf(-iv));   // input gate in [0,1]
  const float og = 1.f / (1.f + expf(-ov));   // output gate
  const float g  = tanhf(gv);                 // candidate

  float num = expf(ig) * g;
  float den = expf(ig);
  const float aw = awi[h];
  #pragma unroll 8
  for (int c = 0; c < CNUM; ++c) {
    const float z  = aw + awh[(size_t)c * H + h];
    const float al = 1.f / (1.f + expf(-z));
    const float e  = expf(al);
    num += e * cinp[(size_t)c * H + h];
    den += e;
  }
  const float c1 = num / den;
  const float h1 = og * tanhf(c1);
  out[h]     = h1;   // h1 first in tuple order
  out[H + h] = c1;
}

// ---------------------------------------------------------------------------
extern "C" void kernel_launch(void* const* d_in, const int* in_sizes, int n_in,
                              void* d_out, int out_size, void* d_ws, size_t ws_size,
                              hipStream_t stream) {
  const float* x     = (const float*)d_in[0];   // input_        (1, 2048)
  const float* cin   = (const float*)d_in[1];   // c_input       (64, 2048)
  const float* h0    = (const float*)d_in[2];   // h0            (1, 2048)
  /* d_in[3] = c0 : unused by the reference */
  const float* Wih   = (const float*)d_in[4];   // weight_ih     (2048, 6144)
  const float* Whh   = (const float*)d_in[5];   // weight_hh     (2048, 6144)
  const float* aWih  = (const float*)d_in[6];   // alpha_weight_ih (2048, 2048)
  const float* aWhh  = (const float*)d_in[7];   // alpha_weight_hh (2048, 2048)
  const float* bias  = (const float*)d_in[8];   // (6144,)
  const float* abias = (const float*)d_in[9];   // (2048,)
  float* out = (float*)d_out;                   // [h1 (2048) | c1 (2048)]
  float* ws  = (float*)d_ws;

  float* preP = ws;                         // 64 * 6144
  float* awiP = preP + NSLICE * H3;         // 64 * 2048
  float* pre  = awiP + NSLICE * H;          // 6144
  float* awi  = pre + H3;                   // 2048
  float* awh  = awi + H;                    // 64 * 2048

  // Bandwidth-dominant matvecs (100 MB + 17 MB of weights, read once, coalesced)
  gate_matvec_partial<<<dim3(H3 / 1024, NSLICE), dim3(256), 0, stream>>>(
      x, h0, Wih, Whh, preP);
  alpha_matvec_partial<<<dim3(H / 1024, NSLICE), dim3(256), 0, stream>>>(
      x, aWih, awiP);
  // 64x2048x2048 GEMM on the f32 WMMA path
  alpha_gemm_wmma<<<dim3(H / 64, CNUM / 16), dim3(128), 0, stream>>>(
      cin, aWhh, awh);
  // Deterministic slice reduction + biases
  finalize_vec<<<dim3((H3 + H) / 256), dim3(256), 0, stream>>>(
      preP, awiP, bias, abias, pre, awi);
  // Gates + exp-normalized attention merge + outputs
  fuse_out<<<dim3(H / 256), dim3(256), 0, stream>>>(pre, awi, awh, cin, out);
}